// SetCriterion_29781303230674
// MI455X (gfx1250) — compile-verified
//
#include <hip/hip_runtime.h>
#include <math.h>

// ---------------------------------------------------------------------------
// Rotation-angle loss for MI455X (gfx1250, wave32).
// ~402 MB streamed -> one scalar. Bandwidth floor ~17us @ 23.3 TB/s.
// Hardware transcendentals (v_sin_f32/v_cos_f32, co-executing TRANS ops) keep
// VALU time (~10us) under the memory floor; accurate-libm sincos would be
// compute-bound (~27us). Per-wave reduction uses V_WMMA_F32_16X16X4_F32.
// ---------------------------------------------------------------------------

typedef float v2f __attribute__((ext_vector_type(2)));
typedef float v8f __attribute__((ext_vector_type(8)));

#define THREADS   256          // 8 wave32s per block
#define NBLOCKS   4096
#define BATCH_B   8192
#define EPS_F     1e-6f

// Sum all 64 values held in (a0,a1) across the 32 lanes of this wave using
// one f32 WMMA.  A is 16x4 (2 VGPRs/lane), B = ones, D[m][n] = rowsum(m).
// Per-lane sum of the 8 D VGPRs covers half the rows; xor-16 combines halves.
// Requires full EXEC (call only from convergent code).
__device__ __forceinline__ float wave_sum64_wmma(float a0, float a1) {
    v2f A;  A[0] = a0;   A[1] = a1;
    v2f Bm; Bm[0] = 1.0f; Bm[1] = 1.0f;
    v8f C = {};
    v8f D = __builtin_amdgcn_wmma_f32_16x16x4_f32(
        /*neg_a=*/false, A, /*neg_b=*/false, Bm,
        /*c_mod=*/(short)0, C, /*reuse_a=*/false, /*reuse_b=*/false);
    float s = ((D[0] + D[1]) + (D[2] + D[3])) + ((D[4] + D[5]) + (D[6] + D[7]));
    s += __shfl_xor(s, 16, 32);
    return s;
}

// R = RZ(yaw) @ RY(pitch) @ RX(roll), 9 entries; hardware sin/cos.
__device__ __forceinline__ void rot_entries(float r, float p, float y, float* m) {
    float sr = __sinf(r), cr = __cosf(r);
    float sp = __sinf(p), cp = __cosf(p);
    float sy = __sinf(y), cy = __cosf(y);
    m[0] = cy * cp;
    m[1] = cy * sp * sr - sy * cr;
    m[2] = cy * sp * cr + sy * sr;
    m[3] = sy * cp;
    m[4] = sy * sp * sr + cy * cr;
    m[5] = sy * sp * cr - cy * sr;
    m[6] = -sp;
    m[7] = cp * sr;
    m[8] = cp * cr;
}

// One pair: angle between the two Euler rotations.
__device__ __forceinline__ float pair_angle(const float* __restrict__ outp,
                                            const float* __restrict__ tgtp,
                                            long long pi) {
    long long base = pi * 6 + 3;   // [.., 1, :] slice of the [2,3] record
    float r1 = __builtin_nontemporal_load(outp + base + 0);
    float p1 = __builtin_nontemporal_load(outp + base + 1);
    float y1 = __builtin_nontemporal_load(outp + base + 2);
    float r2 = __builtin_nontemporal_load(tgtp + base + 0);
    float p2 = __builtin_nontemporal_load(tgtp + base + 1);
    float y2 = __builtin_nontemporal_load(tgtp + base + 2);

    float ma[9], mb[9];
    rot_entries(r1, p1, y1, ma);
    rot_entries(r2, p2, y2, mb);

    float tr = ma[0] * mb[0];
    #pragma unroll
    for (int i = 1; i < 9; ++i) tr = fmaf(ma[i], mb[i], tr);

    float th = 0.5f * (tr - 1.0f);
    th = fminf(fmaxf(th, -1.0f + EPS_F), 1.0f - EPS_F);
    return acosf(th);
}

__device__ __forceinline__ float block_reduce(float a0, float a1) {
    __shared__ float smem[THREADS / 32];
    float w = wave_sum64_wmma(a0, a1);
    int lane = threadIdx.x & 31;
    int wid  = threadIdx.x >> 5;
    if (lane == 0) smem[wid] = w;
    __syncthreads();
    float t = 0.0f;
    if (threadIdx.x == 0) {
        #pragma unroll
        for (int i = 0; i < THREADS / 32; ++i) t += smem[i];
    }
    return t;   // valid in thread 0 only
}

__global__ __launch_bounds__(THREADS)
void rot_loss_partial(const float* __restrict__ outp,
                      const float* __restrict__ tgtp,
                      float* __restrict__ partials,
                      long long npairs) {
    long long gid    = (long long)blockIdx.x * THREADS + threadIdx.x;
    long long stride = (long long)gridDim.x * THREADS;

    float acc0 = 0.0f, acc1 = 0.0f;
    long long pi = gid;
    // Unroll by 2: two independent chains hide TRANS latency.
    for (; pi + stride < npairs; pi += 2 * stride) {
        acc0 += pair_angle(outp, tgtp, pi);
        acc1 += pair_angle(outp, tgtp, pi + stride);
    }
    if (pi < npairs) acc0 += pair_angle(outp, tgtp, pi);

    // All threads convergent here; full 256-thread block -> EXEC all ones.
    float t = block_reduce(acc0, acc1);
    if (threadIdx.x == 0) partials[blockIdx.x] = t;
}

__global__ __launch_bounds__(THREADS)
void rot_loss_final(const float* __restrict__ partials, int n,
                    float* __restrict__ out, float inv_batch) {
    float acc0 = 0.0f, acc1 = 0.0f;
    for (int i = threadIdx.x; i < n; i += 2 * THREADS)           acc0 += partials[i];
    for (int i = threadIdx.x + THREADS; i < n; i += 2 * THREADS) acc1 += partials[i];

    float t = block_reduce(acc0, acc1);
    if (threadIdx.x == 0) out[0] = t * inv_batch;
}

extern "C" void kernel_launch(void* const* d_in, const int* in_sizes, int n_in,
                              void* d_out, int out_size, void* d_ws, size_t ws_size,
                              hipStream_t stream) {
    const float* outp = (const float*)d_in[0];   // [B, NOBJ, 2, 3] f32
    const float* tgtp = (const float*)d_in[1];   // [B, NOBJ, 2, 3] f32
    float* out        = (float*)d_out;           // scalar f32
    float* partials   = (float*)d_ws;

    long long npairs = (long long)in_sizes[0] / 6;   // B * NOBJ

    int nblocks = NBLOCKS;
    size_t need = (size_t)nblocks * sizeof(float);
    if (ws_size < need) {
        nblocks = (int)(ws_size / sizeof(float));
        if (nblocks < 1) nblocks = 1;
    }

    rot_loss_partial<<<nblocks, THREADS, 0, stream>>>(outp, tgtp, partials, npairs);
    rot_loss_final<<<1, THREADS, 0, stream>>>(partials, nblocks, out,
                                              1.0f / (float)BATCH_B);
}